// Pixelated_28613072126046
// MI455X (gfx1250) — compile-verified
//
#include <hip/hip_runtime.h>

// Bilinear "interp2d with zeros padding" gather kernel for MI455X (gfx1250).
//
// Memory strategy (the only thing that matters here — the kernel is gather/
// bandwidth bound, ~11us roofline at 23.3 TB/s):
//   * coords (x,y) and output are single-touch streams  -> non-temporal
//     loads/stores (CDNA5 TH=NT) so they do not evict the image from L2.
//   * the 64MB image fits in the 192MB L2 -> default (RT) temporal hint on
//     the gathers keeps it resident; gathers mostly hit L2.
//   * global_prefetch_b8 (speculative, fault-safe) pulls the coord stream
//     into GL2 ahead of the wave.
//   * the two horizontal taps of each row are adjacent -> one 8-byte load
//     per row (2 gather loads per point instead of 4).
//   * out-of-bounds lanes (~17%) branch around the gathers entirely; VMEM
//     traffic is per-active-lane under EXEC, so this saves real bandwidth.

typedef float f4 __attribute__((ext_vector_type(4)));

// 8-byte pair with only 4-byte alignment guarantee (x0 index may be odd).
// gfx1250 DWORD alignment mode handles dword-aligned b64 loads.
struct __attribute__((packed, aligned(4))) F2 {
    float lo, hi;
};

struct InterpParams {
    float x0, y0, scale;
    float nx, ny;        // xn = (x - x0) * nx, with nx = 2 / fov_x
};

#define IMG_W 4096
#define IMG_H 4096

__device__ __forceinline__ float interp_one(float xq, float yq,
                                            const float* __restrict__ img,
                                            const InterpParams& p) {
    float xn = (xq - p.x0) * p.nx;
    float yn = (yq - p.y0) * p.ny;
    float res = 0.0f;
    // zeros padding: anything with |xn|>1 or |yn|>1 contributes 0 and,
    // crucially, issues no gather traffic for those lanes.
    if (!(xn < -1.0f || xn > 1.0f || yn < -1.0f || yn > 1.0f)) {
        // pixel-space coords, exactly as reference
        float xi = 0.5f * ((xn + 1.0f) * (float)IMG_W - 1.0f);
        float yi = 0.5f * ((yn + 1.0f) * (float)IMG_H - 1.0f);
        int x0i = (int)floorf(xi);
        int y0i = (int)floorf(yi);
        x0i = min(max(x0i, 0), IMG_W - 2);
        y0i = min(max(y0i, 0), IMG_H - 2);
        // weights from the *clamped* integer corners (matches reference)
        float dx0 = xi - (float)x0i;
        float dx1 = (float)(x0i + 1) - xi;
        float dy0 = yi - (float)y0i;
        float dy1 = (float)(y0i + 1) - yi;
        int base = y0i * IMG_W + x0i;
        // two taps per row are contiguous -> one b64 gather per row
        F2 top = *reinterpret_cast<const F2*>(img + base);          // fa, fc
        F2 bot = *reinterpret_cast<const F2*>(img + base + IMG_W);  // fb, fd
        float row0 = top.lo * dx1 + top.hi * dx0;  // y0 row
        float row1 = bot.lo * dx1 + bot.hi * dx0;  // y1 row
        res = p.scale * (row0 * dy1 + row1 * dy0);
    }
    return res;
}

__global__ __launch_bounds__(256)
void Pixelated_bilerp_kernel(const float* __restrict__ x,
                             const float* __restrict__ y,
                             const float* __restrict__ sx0,
                             const float* __restrict__ sy0,
                             const float* __restrict__ img,
                             const float* __restrict__ sps,
                             const float* __restrict__ sscale,
                             float* __restrict__ out,
                             int n) {
    const int gid = blockIdx.x * blockDim.x + threadIdx.x;
    const int i0 = gid * 4;
    if (i0 >= n) return;

    InterpParams p;
    p.x0 = *sx0;
    p.y0 = *sy0;
    p.scale = *sscale;
    const float ps = *sps;
    p.nx = 2.0f / (ps * (float)IMG_W);  // == 2 / fov_x
    p.ny = 2.0f / (ps * (float)IMG_H);

    // Speculative prefetch of the coordinate streams ~64KB ahead into GL2.
    // gfx1250 global_prefetch_b8; translation faults are silently dropped.
    __builtin_prefetch(x + i0 + 16384, 0, 0);
    __builtin_prefetch(y + i0 + 16384, 0, 0);

    if (i0 + 3 < n) {
        // Hot path: one float4 group per thread, non-temporal streams.
        const f4 xq = __builtin_nontemporal_load(reinterpret_cast<const f4*>(x + i0));
        const f4 yq = __builtin_nontemporal_load(reinterpret_cast<const f4*>(y + i0));
        f4 r;
#pragma unroll
        for (int k = 0; k < 4; ++k) {
            r[k] = interp_one(xq[k], yq[k], img, p);
        }
        __builtin_nontemporal_store(r, reinterpret_cast<f4*>(out + i0));
    } else {
        // Tail (never taken for 4096x4096, kept for correctness)
        for (int i = i0; i < n; ++i) {
            float v = interp_one(x[i], y[i], img, p);
            __builtin_nontemporal_store(v, out + i);
        }
    }
}

extern "C" void kernel_launch(void* const* d_in, const int* in_sizes, int n_in,
                              void* d_out, int out_size, void* d_ws, size_t ws_size,
                              hipStream_t stream) {
    (void)n_in; (void)d_ws; (void)ws_size; (void)out_size;
    const float* x      = (const float*)d_in[0];
    const float* y      = (const float*)d_in[1];
    const float* sx0    = (const float*)d_in[2];
    const float* sy0    = (const float*)d_in[3];
    const float* img    = (const float*)d_in[4];
    const float* sps    = (const float*)d_in[5];
    const float* sscale = (const float*)d_in[6];
    float* out = (float*)d_out;

    const int n = in_sizes[0];                 // 4096*4096 query points
    const int n_threads = (n + 3) / 4;         // 4 points per thread
    const int block = 256;                     // 8 wave32 per block
    const int grid = (n_threads + block - 1) / block;

    Pixelated_bilerp_kernel<<<grid, block, 0, stream>>>(
        x, y, sx0, sy0, img, sps, sscale, out, n);
}